// GAT_TRANSFORMER_30760555773966
// MI455X (gfx1250) — compile-verified
//
#include <hip/hip_runtime.h>
#include <hip/hip_bf16.h>
#include <math.h>

// =====================================================================
// SuperGlue-style matcher for MI455X (gfx1250, wave32, WMMA).
// Dense GEMMs use V_WMMA_F32_16X16X4_F32 (fp32 in, fp32 accum) with
// compile-time K (branch-free K loop) and clamped-address edge handling
// so the inner loop is pure loads + wmma (no EXEC juggling).
// Sinkhorn (100 iters) runs LDS-resident, one workgroup per batch.
// =====================================================================

#define BSZ 512
#define NN  63
#define MM  128
#define DD  128

typedef __attribute__((ext_vector_type(2))) float v2f;
typedef __attribute__((ext_vector_type(8))) float v8f;

// ---------------------------------------------------------------------
// Generic conv1x1 GEMM:  Y[b,o,l] (+)= sum_k W[o, cOff+k] * X[b,k,l] (+ bias[o])
// One wave per 16x(16*NT) output strip; K-loop (compile-time) in steps of 4.
//   A frag (16x4 f32): lanes 0-15 -> M=lane, K={0,1}; lanes 16-31 -> K={2,3}
//   B frag (4x16 f32): lanes 0-15 -> N=lane, K={0,1}; lanes 16-31 -> K={2,3}
//   C/D (16x16 f32):   vgpr r -> M=r (lanes<16) / M=r+8, N=lane&15
// Edge columns load from clamped addresses (their D columns are never
// stored, so garbage contributions are harmless). K guards removed by
// padding (K always multiple of 4).
// flags: bit0 = add bias, bit1 = accumulate into existing Y
// ---------------------------------------------------------------------
template<int K, int NT>
__global__ void gemm_wmma_k(const float* __restrict__ X, const float* __restrict__ W,
                            const float* __restrict__ bias, float* __restrict__ Y,
                            int Cout, int L, int ldW, int cOff,
                            int nOT, int nGT, int flags)
{
    int blk = blockIdx.x;
    int gt = blk % nGT; blk /= nGT;
    int ot = blk % nOT; blk /= nOT;
    int b  = blk;
    int lane = threadIdx.x;
    int half = lane >> 4;      // 0 or 1
    int lidx = lane & 15;
    int o0 = ot * 16;
    int l0 = gt * (16 * NT);

    const float* Xb = X + (size_t)b * K * L;
    float* Yb = Y + (size_t)b * Cout * L;

    int lc[NT], lcl[NT];
#pragma unroll
    for (int j = 0; j < NT; ++j) {
        lc[j]  = l0 + j * 16 + lidx;
        lcl[j] = (lc[j] < L) ? lc[j] : (L - 1);   // clamped, always in-bounds
    }

    v8f c[NT];
    if (flags & 2) {
#pragma unroll
        for (int j = 0; j < NT; ++j)
#pragma unroll
            for (int r = 0; r < 8; ++r)
                c[j][r] = Yb[(size_t)(o0 + r + half * 8) * L + lcl[j]];
    } else {
#pragma unroll
        for (int j = 0; j < NT; ++j)
#pragma unroll
            for (int r = 0; r < 8; ++r) c[j][r] = 0.f;
    }

    const float* Wr = W + (size_t)(o0 + lidx) * ldW + cOff;   // A row for this lane
    int kq = half * 2;
#pragma unroll 4
    for (int k0 = 0; k0 < K; k0 += 4) {
        int ka = k0 + kq;
        v2f a;
        a.x = Wr[ka];
        a.y = Wr[ka + 1];
#pragma unroll
        for (int j = 0; j < NT; ++j) {
            v2f bf;
            bf.x = Xb[(size_t)ka * L + lcl[j]];
            bf.y = Xb[(size_t)(ka + 1) * L + lcl[j]];
            c[j] = __builtin_amdgcn_wmma_f32_16x16x4_f32(false, a, false, bf,
                                                         (short)0, c[j], false, false);
        }
    }

#pragma unroll
    for (int j = 0; j < NT; ++j) {
        if (lc[j] < L) {
#pragma unroll
            for (int r = 0; r < 8; ++r) {
                int oo = o0 + r + half * 8;
                float vout = c[j][r];
                if (flags & 1) vout += bias[oo];
                Yb[(size_t)oo * L + lc[j]] = vout;
            }
        }
    }
}

// ---------------------------------------------------------------------
// sim[b,n,m] = sum_d d0[b,d,n] * d1[b,d,m]  (+ -inf where cam lost)
// One wave per 16x64 strip of the 63x128 output; A rows >= 63 load from
// clamped addresses (their D rows are never stored).
// ---------------------------------------------------------------------
__global__ void sim_wmma_k(const float* __restrict__ d0, const float* __restrict__ d1,
                           const float* __restrict__ lost, float* __restrict__ sim)
{
    const int NT = 4;
    int blk = blockIdx.x;
    int mg = blk % 2; blk /= 2;           // m-group (2 groups of 4 tiles)
    int nt = blk % 4; blk /= 4;           // n-tile
    int b  = blk;
    int lane = threadIdx.x;
    int half = lane >> 4, lidx = lane & 15;
    int n0 = nt * 16, m0 = mg * 64;

    const float* A  = d0 + (size_t)b * DD * NN;
    const float* Bp = d1 + (size_t)b * DD * MM;

    int na  = n0 + lidx;
    int nacl = (na < NN) ? na : (NN - 1);
    int mb[NT];
#pragma unroll
    for (int j = 0; j < NT; ++j) mb[j] = m0 + j * 16 + lidx;

    v8f c[NT];
#pragma unroll
    for (int j = 0; j < NT; ++j)
#pragma unroll
        for (int r = 0; r < 8; ++r) c[j][r] = 0.f;

    int kq = half * 2;
#pragma unroll 4
    for (int k0 = 0; k0 < DD; k0 += 4) {
        int ka = k0 + kq;
        v2f a;
        a.x = A[(size_t)ka * NN + nacl];
        a.y = A[(size_t)(ka + 1) * NN + nacl];
#pragma unroll
        for (int j = 0; j < NT; ++j) {
            v2f bf;
            bf.x = Bp[(size_t)ka * MM + mb[j]];
            bf.y = Bp[(size_t)(ka + 1) * MM + mb[j]];
            c[j] = __builtin_amdgcn_wmma_f32_16x16x4_f32(false, a, false, bf,
                                                         (short)0, c[j], false, false);
        }
    }

    float* S = sim + (size_t)b * NN * MM;
#pragma unroll
    for (int j = 0; j < NT; ++j) {
        float biasv = (lost[(size_t)b * MM + mb[j]] > 0.5f) ? -INFINITY : 0.f;
#pragma unroll
        for (int r = 0; r < 8; ++r) {
            int n = n0 + r + half * 8;
            if (n < NN) S[(size_t)n * MM + mb[j]] = c[j][r] + biasv;
        }
    }
}

// ---------------------------------------------------------------------
// BatchNorm (training mode): stats over (batch, length) per channel.
// ---------------------------------------------------------------------
__global__ void bn_stats_k(const float* __restrict__ X, float* __restrict__ stats,
                           int C, int L)
{
    int c = blockIdx.x;
    int t = threadIdx.x;
    int NL = BSZ * L;
    float s = 0.f, sq = 0.f;
    for (int i = t; i < NL; i += 256) {
        int b = i / L, l = i - b * L;
        float v = X[(size_t)b * C * L + (size_t)c * L + l];
        s += v; sq += v * v;
    }
    __shared__ float sh[512];
    sh[t] = s; sh[256 + t] = sq;
    __syncthreads();
    for (int o = 128; o > 0; o >>= 1) {
        if (t < o) { sh[t] += sh[t + o]; sh[256 + t] += sh[256 + t + o]; }
        __syncthreads();
    }
    if (t == 0) {
        float mean = sh[0] / (float)NL;
        float var  = sh[256] / (float)NL - mean * mean;
        stats[c * 2]     = mean;
        stats[c * 2 + 1] = rsqrtf(fmaxf(var, 0.f) + 1e-5f);
    }
}

__global__ void bn_apply_k(float* __restrict__ X, const float* __restrict__ stats,
                           const float* __restrict__ g, const float* __restrict__ beta,
                           int C, int L)
{
    size_t i = (size_t)blockIdx.x * blockDim.x + threadIdx.x;
    size_t tot = (size_t)BSZ * C * L;
    if (i >= tot) return;
    int c = (int)((i / (size_t)L) % (size_t)C);
    float v = X[i];
    v = g[c] * (v - stats[c * 2]) * stats[c * 2 + 1] + beta[c];
    X[i] = fmaxf(v, 0.f);
}

// ---------------------------------------------------------------------
// Fused MHA (flash-style online softmax). One workgroup per (batch, head).
// Channel layout from reshape(B, HD=32, H=4, L): channel = hd*4 + h.
// Writes attention output in-place into the q buffer.
// ---------------------------------------------------------------------
__global__ void attn_k(float* __restrict__ q, const float* __restrict__ k,
                       const float* __restrict__ v, int L)
{
    int h = blockIdx.x & 3;
    int b = blockIdx.x >> 2;
    __shared__ float ksh[32 * 128];
    __shared__ float vsh[32 * 128];
    size_t base = (size_t)b * DD * L;
    int t = threadIdx.x;
    for (int i = t; i < 32 * L; i += blockDim.x) {
        int hd = i / L, m = i - hd * L;
        ksh[i] = k[base + (size_t)(hd * 4 + h) * L + m];
        vsh[i] = v[base + (size_t)(hd * 4 + h) * L + m];
    }
    __syncthreads();
    int n = t;
    if (n < L) {
        const float scale = 0.17677669529663687f;  // 1/sqrt(32)
        float qr[32];
#pragma unroll
        for (int hd = 0; hd < 32; ++hd)
            qr[hd] = q[base + (size_t)(hd * 4 + h) * L + n] * scale;
        float mx = -INFINITY, sum = 0.f;
        float acc[32];
#pragma unroll
        for (int hd = 0; hd < 32; ++hd) acc[hd] = 0.f;
        for (int m = 0; m < L; ++m) {
            float s = 0.f;
#pragma unroll
            for (int hd = 0; hd < 32; ++hd) s += qr[hd] * ksh[hd * L + m];
            float nm = fmaxf(mx, s);
            float corr = expf(mx - nm);   // first iter: exp(-inf)=0
            float e = expf(s - nm);
            sum = sum * corr + e;
#pragma unroll
            for (int hd = 0; hd < 32; ++hd)
                acc[hd] = acc[hd] * corr + e * vsh[hd * L + m];
            mx = nm;
        }
        float inv = 1.f / sum;
#pragma unroll
        for (int hd = 0; hd < 32; ++hd)
            q[base + (size_t)(hd * 4 + h) * L + n] = acc[hd] * inv;
    }
}

// ---------------------------------------------------------------------
// Input prep: prior_pos/dir (4-channel padded, transposed), cam_dir,
// lost mask, and zero-padded kenc layer-0 weight (32x3 -> 32x4).
// ---------------------------------------------------------------------
__global__ void prep_k(const float* __restrict__ feat, const float* __restrict__ cam,
                       const float* __restrict__ kw0,
                       float* priorPos, float* dir0T, float* od,
                       float* camdirB, float* dir1T, float* lost, float* w0pad)
{
    int i = blockIdx.x * blockDim.x + threadIdx.x;
    const int totN = BSZ * NN;
    const int totM = BSZ * MM;
    if (i < 128) {   // pad 32x3 weight to 32x4 (zero 4th column)
        int o = i >> 2, c = i & 3;
        w0pad[i] = (c < 3) ? kw0[o * 3 + c] : 0.f;
    }
    if (i < totN) {
        int b = i / NN, n = i - b * NN;
        const float* fr = feat + (size_t)i * 8;
        float x = fr[0], y = fr[1], z = fr[2];
        priorPos[i * 3 + 0] = x; priorPos[i * 3 + 1] = y; priorPos[i * 3 + 2] = z;
        od[i] = fr[7];
        float nm = sqrtf(x * x + y * y + z * z);
        float inv = 1.f / fmaxf(nm, 1e-12f);
        float* dt = dir0T + (size_t)b * 4 * NN;
        dt[0 * NN + n] = x * inv; dt[1 * NN + n] = y * inv;
        dt[2 * NN + n] = z * inv; dt[3 * NN + n] = 0.f;
    } else if (i < totN + totM) {
        int j = i - totN;
        int b = j / MM, m = j - b * MM;
        const float* cr = cam + (size_t)j * 3;
        float x = cr[0], y = cr[1], z = cr[2];
        float nm = sqrtf(x * x + y * y + z * z);
        lost[j] = (nm < 1e-4f) ? 1.f : 0.f;
        float inv = 1.f / fmaxf(nm, 1e-12f);
        camdirB[j * 3 + 0] = x * inv; camdirB[j * 3 + 1] = y * inv; camdirB[j * 3 + 2] = z * inv;
        float* dt = dir1T + (size_t)b * 4 * MM;
        dt[0 * MM + m] = x * inv; dt[1 * MM + m] = y * inv;
        dt[2 * MM + m] = z * inv; dt[3 * MM + m] = 0.f;
    }
}

// ---------------------------------------------------------------------
// Row softmax (prob output) + attn_dir + var, one block per (b,n).
// ---------------------------------------------------------------------
__global__ void softvar_k(const float* __restrict__ sim, const float* __restrict__ camdirB,
                          float* __restrict__ probOut, float* __restrict__ varb)
{
    __shared__ float sh[128];
    int n = blockIdx.x % NN;
    int b = blockIdx.x / NN;
    int m = threadIdx.x;
    const float* row = sim + ((size_t)(b * NN + n)) * MM;
    float s = row[m];
    sh[m] = s; __syncthreads();
    for (int o = 64; o > 0; o >>= 1) { if (m < o) sh[m] = fmaxf(sh[m], sh[m + o]); __syncthreads(); }
    float mx = sh[0]; __syncthreads();
    float e = (s == -INFINITY) ? 0.f : expf(s - mx);
    sh[m] = e; __syncthreads();
    for (int o = 64; o > 0; o >>= 1) { if (m < o) sh[m] += sh[m + o]; __syncthreads(); }
    float p = e / sh[0]; __syncthreads();
    probOut[((size_t)(b * NN + n)) * MM + m] = p;

    const float* cd = camdirB + ((size_t)b * MM + m) * 3;
    float c0 = cd[0], c1 = cd[1], c2 = cd[2];
    float a0, a1, a2;
    sh[m] = p * c0; __syncthreads();
    for (int o = 64; o > 0; o >>= 1) { if (m < o) sh[m] += sh[m + o]; __syncthreads(); }
    a0 = sh[0]; __syncthreads();
    sh[m] = p * c1; __syncthreads();
    for (int o = 64; o > 0; o >>= 1) { if (m < o) sh[m] += sh[m + o]; __syncthreads(); }
    a1 = sh[0]; __syncthreads();
    sh[m] = p * c2; __syncthreads();
    for (int o = 64; o > 0; o >>= 1) { if (m < o) sh[m] += sh[m + o]; __syncthreads(); }
    a2 = sh[0]; __syncthreads();
    float nrm = sqrtf(a0 * a0 + a1 * a1 + a2 * a2);
    float inv = 1.f / fmaxf(nrm, 1e-12f);
    a0 *= inv; a1 *= inv; a2 *= inv;
    float dx = c0 - a0, dy = c1 - a1, dz = c2 - a2;
    float g = sqrtf(dx * dx + dy * dy + dz * dz);
    sh[m] = p * g; __syncthreads();
    for (int o = 64; o > 0; o >>= 1) { if (m < o) sh[m] += sh[m + o]; __syncthreads(); }
    if (m == 0) varb[b * NN + n] = sh[0];
}

// ---------------------------------------------------------------------
// Sinkhorn (100 iters) fully LDS-resident, plus fused matching.
// One workgroup (256 threads) per batch; Z is 64x129 fp32 (33KB LDS).
// ---------------------------------------------------------------------
__global__ void sinkhorn_k(const float* __restrict__ sim, const float* __restrict__ alpha_p,
                           float* __restrict__ scoresOut, float* __restrict__ indicesOut,
                           float* __restrict__ mscoresW, int* __restrict__ idx0W)
{
    int b = blockIdx.x;
    int t = threadIdx.x;
    __shared__ float Z[64 * 129];
    __shared__ float uu[64], vv[129];
    __shared__ int   idx0s[NN]; __shared__ float max0s[NN]; __shared__ int idx1s[MM];

    float alpha = alpha_p[0];
    const float rD = 0.08838834764831845f;   // 1/sqrt(128)
    const float* S = sim + (size_t)b * NN * MM;
    for (int i = t; i < 64 * 129; i += 256) {
        int n = i / 129, m = i - n * 129;
        Z[i] = (n < NN && m < MM) ? S[(size_t)n * MM + m] * rD : alpha;
    }
    if (t < 64)  uu[t] = 0.f;
    if (t < 129) vv[t] = 0.f;
    __syncthreads();

    const float normv    = -logf(191.f);
    const float lmu_last = logf(128.f) + normv;
    const float lnu_last = logf(63.f)  + normv;

    for (int it = 0; it < 100; ++it) {
        if (t < 64) {
            float mx = -INFINITY;
            for (int m = 0; m < 129; ++m) mx = fmaxf(mx, Z[t * 129 + m] + vv[m]);
            float s = 0.f;
            for (int m = 0; m < 129; ++m) s += expf(Z[t * 129 + m] + vv[m] - mx);
            uu[t] = ((t < NN) ? normv : lmu_last) - (mx + logf(s));
        }
        __syncthreads();
        if (t < 129) {
            float mx = -INFINITY;
            for (int n = 0; n < 64; ++n) mx = fmaxf(mx, Z[n * 129 + t] + uu[n]);
            float s = 0.f;
            for (int n = 0; n < 64; ++n) s += expf(Z[n * 129 + t] + uu[n] - mx);
            vv[t] = ((t < MM) ? normv : lnu_last) - (mx + logf(s));
        }
        __syncthreads();
    }

    float* SO = scoresOut + (size_t)b * 64 * 129;
    for (int i = t; i < 64 * 129; i += 256) {
        int n = i / 129, m = i - n * 129;
        float sc = Z[i] + uu[n] + vv[m] - normv;
        SO[i] = sc;
        Z[i]  = sc;
    }
    __syncthreads();

    // matching on inner 63x128 block
    if (t < NN) {
        float mx = -INFINITY; int am = 0;
        for (int m = 0; m < MM; ++m) {
            float zz = Z[t * 129 + m];
            if (zz > mx) { mx = zz; am = m; }
        }
        idx0s[t] = am; max0s[t] = mx;
    }
    if (t >= 64 && t < 64 + MM) {
        int m = t - 64;
        float mx = -INFINITY; int an = 0;
        for (int n = 0; n < NN; ++n) {
            float zz = Z[n * 129 + m];
            if (zz > mx) { mx = zz; an = n; }
        }
        idx1s[m] = an;
    }
    __syncthreads();
    if (t < NN) {
        int i0 = idx0s[t];
        bool mutual = (idx1s[i0] == t);
        float ms = mutual ? expf(max0s[t]) : 0.f;
        bool valid = mutual && (ms > 0.6f);
        indicesOut[(size_t)b * NN + t] = (float)(valid ? i0 : -1);
        mscoresW[(size_t)b * NN + t] = ms;
        idx0W[(size_t)b * NN + t] = i0;
    }
}

// ---------------------------------------------------------------------
// Final tiny MLP heads (pos_pred and cov_pred), one thread per (b,n).
// ---------------------------------------------------------------------
__global__ void heads_k(const float* __restrict__ priorPos, const float* __restrict__ od,
                        const float* __restrict__ camdirB, const int* __restrict__ idx0W,
                        const float* __restrict__ mscoresW, const float* __restrict__ varb,
                        const float* p0w, const float* p0b, const float* p1w, const float* p1b,
                        const float* p2w, const float* p2b, const float* p3w, const float* p3b,
                        const float* c0w, const float* c0b, const float* c1w, const float* c1b,
                        const float* c2w, const float* c2b,
                        float* __restrict__ posOut, float* __restrict__ covOut)
{
    int i = blockIdx.x * blockDim.x + threadIdx.x;
    if (i >= BSZ * NN) return;
    int b = i / NN;
    float f[8];
    f[0] = priorPos[i * 3 + 0]; f[1] = priorPos[i * 3 + 1]; f[2] = priorPos[i * 3 + 2];
    int m = idx0W[i];
    const float* cd = camdirB + ((size_t)b * MM + m) * 3;
    float d = od[i];
    f[3] = d * cd[0]; f[4] = d * cd[1]; f[5] = d * cd[2];
    f[6] = varb[i]; f[7] = mscoresW[i];

    float hA[64], hB[64];
    // pos_pred: 8->32->64->32->3 (relu,relu,relu,none)
    for (int o = 0; o < 32; ++o) { float s = p0b[o]; for (int c = 0; c < 8;  ++c) s += p0w[o * 8  + c] * f[c];  hA[o] = fmaxf(s, 0.f); }
    for (int o = 0; o < 64; ++o) { float s = p1b[o]; for (int c = 0; c < 32; ++c) s += p1w[o * 32 + c] * hA[c]; hB[o] = fmaxf(s, 0.f); }
    for (int o = 0; o < 32; ++o) { float s = p2b[o]; for (int c = 0; c < 64; ++c) s += p2w[o * 64 + c] * hB[c]; hA[o] = fmaxf(s, 0.f); }
    for (int o = 0; o < 3;  ++o) { float s = p3b[o]; for (int c = 0; c < 32; ++c) s += p3w[o * 32 + c] * hA[c]; posOut[i * 3 + o] = f[o] + s; }
    // cov_pred: 8->32->16->1 (relu,relu,sigmoid) then clip to [1e-4, 10]
    for (int o = 0; o < 32; ++o) { float s = c0b[o]; for (int c = 0; c < 8;  ++c) s += c0w[o * 8  + c] * f[c];  hA[o] = fmaxf(s, 0.f); }
    for (int o = 0; o < 16; ++o) { float s = c1b[o]; for (int c = 0; c < 32; ++c) s += c1w[o * 32 + c] * hA[c]; hB[o] = fmaxf(s, 0.f); }
    {
        float s = c2b[0];
        for (int c = 0; c < 16; ++c) s += c2w[c] * hB[c];
        s = 1.f / (1.f + expf(-s));
        covOut[i] = fminf(fmaxf(s, 1e-4f), 10.f);
    }
}

// =====================================================================
// Host: parameter pytree (JAX dict flattening = sorted keys) and launch.
// =====================================================================
extern "C" void kernel_launch(void* const* d_in, const int* in_sizes, int n_in,
                              void* d_out, int out_size, void* d_ws, size_t ws_size,
                              hipStream_t stream)
{
    (void)in_sizes; (void)n_in; (void)out_size; (void)ws_size;
    auto F = [&](int i) -> const float* { return (const float*)d_in[i]; };

    // ---- params (sorted-key pytree order) ----
    const float* bin_score = F(0);
    const float* c0b = F(1); const float* c0w = F(2);
    const float* c1b = F(3); const float* c1w = F(4);
    const float* c2b = F(5); const float* c2w = F(6);
    struct GL { const float *b1,*b2,*beta1,*bk,*bm,*bq,*bv,*g1,*w1,*w2,*wk,*wm,*wq,*wv; };
    GL gl[4];
    for (int l = 0; l < 4; ++l) {
        int base = 7 + l * 14;
        gl[l].b1 = F(base + 0);  gl[l].b2 = F(base + 1);  gl[l].beta1 = F(base + 2);
        gl[l].bk = F(base + 3);  gl[l].bm = F(base + 4);  gl[l].bq = F(base + 5);
        gl[l].bv = F(base + 6);  gl[l].g1 = F(base + 7);  gl[l].w1 = F(base + 8);
        gl[l].w2 = F(base + 9);  gl[l].wk = F(base + 10); gl[l].wm = F(base + 11);
        gl[l].wq = F(base + 12); gl[l].wv = F(base + 13);
    }
    const float* ke_b[4];  const float* ke_beta[3]; const float* ke_g[3]; const float* ke_w[4];
    ke_b[0] = F(63); ke_beta[0] = F(64); ke_g[0] = F(65); ke_w[0] = F(66);
    ke_b[1] = F(67); ke_beta[1] = F(68); ke_g[1] = F(69); ke_w[1] = F(70);
    ke_b[2] = F(71); ke_beta[2] = F(72); ke_g[2] = F(73); ke_w[2] = F(74);
    ke_b[3] = F(75); ke_w[3] = F(76);
    const float* p0b = F(77); const float* p0w = F(78);
    const float* p1b = F(79); const float* p1w = F(80);
    const float* p2b = F(81); const float* p2w = F(82);
    const float* p3b = F(83); const float* p3w = F(84);
    const float* feat = F(85); const float* cam = F(86);

    // ---- workspace layout (floats) ----
    float* wsf = (float*)d_ws;
    size_t off = 0;
    auto A_ = [&](size_t n) { size_t r = off; off += n; return r; };
    size_t o_qb   = A_((size_t)BSZ * 128 * 128);   // also kenc ping buffer
    size_t o_kb   = A_((size_t)BSZ * 128 * 128);   // also kenc pong buffer
    size_t o_vb   = A_((size_t)BSZ * 128 * 128);
    size_t o_msg  = A_((size_t)BSZ * 128 * 128);
    size_t o_hpre = A_((size_t)BSZ * 256 * 128);   // sim aliases this after GNN
    size_t o_d0   = A_((size_t)BSZ * 128 * NN);
    size_t o_d1   = A_((size_t)BSZ * 128 * MM);
    size_t o_dir0 = A_((size_t)BSZ * 4 * NN);
    size_t o_dir1 = A_((size_t)BSZ * 4 * MM);
    size_t o_pp   = A_((size_t)BSZ * NN * 3);
    size_t o_cd   = A_((size_t)BSZ * MM * 3);
    size_t o_od   = A_((size_t)BSZ * NN);
    size_t o_lost = A_((size_t)BSZ * MM);
    size_t o_var  = A_((size_t)BSZ * NN);
    size_t o_ms   = A_((size_t)BSZ * NN);
    size_t o_idx  = A_((size_t)BSZ * NN);
    size_t o_st   = A_(512);
    size_t o_w0p  = A_(128);

    float* qb   = wsf + o_qb;   float* kb = wsf + o_kb;   float* vb = wsf + o_vb;
    float* msgb = wsf + o_msg;  float* hpre = wsf + o_hpre;
    float* d0p  = wsf + o_d0;   float* d1p = wsf + o_d1;
    float* simb = hpre;                          // alias: hpre dead after GNN
    float* dir0T = wsf + o_dir0; float* dir1T = wsf + o_dir1;
    float* priorPos = wsf + o_pp; float* camdirB = wsf + o_cd;
    float* odb = wsf + o_od;    float* lostb = wsf + o_lost;
    float* varb = wsf + o_var;  float* msW = wsf + o_ms;
    int*   idx0W = (int*)(wsf + o_idx);
    float* statsb = wsf + o_st; float* w0pad = wsf + o_w0p;

    // ---- output layout ----
    float* outp = (float*)d_out;
    float* probOut   = outp;
    float* posOut    = probOut + (size_t)BSZ * NN * MM;
    float* covOut    = posOut + (size_t)BSZ * NN * 3;
    float* scoresOut = covOut + (size_t)BSZ * NN;
    float* indOut    = scoresOut + (size_t)BSZ * 64 * 129;

    auto gemm = [&](const float* X, const float* W, const float* bias, float* Y,
                    int K, int Cout, int L, int ldW, int cOff, int flags) {
        int nOT = Cout / 16;
        int nLT = (L + 15) / 16;
        int nGT = (nLT + 3) / 4;
        int grid = BSZ * nOT * nGT;
        switch (K) {
        case 4:   gemm_wmma_k<4,  4><<<grid, 32, 0, stream>>>(X, W, bias, Y, Cout, L, ldW, cOff, nOT, nGT, flags); break;
        case 32:  gemm_wmma_k<32, 4><<<grid, 32, 0, stream>>>(X, W, bias, Y, Cout, L, ldW, cOff, nOT, nGT, flags); break;
        case 64:  gemm_wmma_k<64, 4><<<grid, 32, 0, stream>>>(X, W, bias, Y, Cout, L, ldW, cOff, nOT, nGT, flags); break;
        case 128: gemm_wmma_k<128,4><<<grid, 32, 0, stream>>>(X, W, bias, Y, Cout, L, ldW, cOff, nOT, nGT, flags); break;
        case 256: gemm_wmma_k<256,4><<<grid, 32, 0, stream>>>(X, W, bias, Y, Cout, L, ldW, cOff, nOT, nGT, flags); break;
        default:  break;
        }
    };
    auto bn = [&](float* Xp, const float* g, const float* beta, int C, int L) {
        bn_stats_k<<<C, 256, 0, stream>>>(Xp, statsb, C, L);
        size_t tot = (size_t)BSZ * C * L;
        bn_apply_k<<<(int)((tot + 255) / 256), 256, 0, stream>>>(Xp, statsb, g, beta, C, L);
    };

    // 0) prep (also pads kenc layer-0 weight to 32x4)
    {
        int tot = BSZ * (NN + MM);
        prep_k<<<(tot + 255) / 256, 256, 0, stream>>>(feat, cam, ke_w[0], priorPos, dir0T,
                                                      odb, camdirB, dir1T, lostb, w0pad);
    }

    // 1) keypoint encoder for both streams (separate BN stats per stream)
    for (int s = 0; s < 2; ++s) {
        int L = s ? MM : NN;
        const float* dirT = s ? dir1T : dir0T;
        float* dS = s ? d1p : d0p;
        gemm(dirT, w0pad, ke_b[0], qb, 4, 32, L, 4, 0, 1);
        bn(qb, ke_g[0], ke_beta[0], 32, L);
        gemm(qb, ke_w[1], ke_b[1], kb, 32, 64, L, 32, 0, 1);
        bn(kb, ke_g[1], ke_beta[1], 64, L);
        gemm(kb, ke_w[2], ke_b[2], qb, 64, 64, L, 64, 0, 1);
        bn(qb, ke_g[2], ke_beta[2], 64, L);
        gemm(qb, ke_w[3], ke_b[3], dS, 64, 128, L, 64, 0, 1);
    }

    // 2) GNN: 4 layers x 2 streams of self-attention + MLP (BN training mode)
    for (int l = 0; l < 4; ++l) {
        for (int s = 0; s < 2; ++s) {
            float* x = s ? d1p : d0p;
            int L = s ? MM : NN;
            gemm(x, gl[l].wq, gl[l].bq, qb, 128, 128, L, 128, 0, 1);
            gemm(x, gl[l].wk, gl[l].bk, kb, 128, 128, L, 128, 0, 1);
            gemm(x, gl[l].wv, gl[l].bv, vb, 128, 128, L, 128, 0, 1);
            attn_k<<<BSZ * 4, 128, 0, stream>>>(qb, kb, vb, L);     // out -> qb
            gemm(qb, gl[l].wm, gl[l].bm, msgb, 128, 128, L, 128, 0, 1);
            // h = W1[:, :128] x + W1[:, 128:] msg + b1
            gemm(x,    gl[l].w1, gl[l].b1, hpre, 128, 256, L, 256, 0,   1);
            gemm(msgb, gl[l].w1, nullptr,  hpre, 128, 256, L, 256, 128, 2);
            bn(hpre, gl[l].g1, gl[l].beta1, 256, L);
            // x += W2 h + b2 (residual via accumulate)
            gemm(hpre, gl[l].w2, gl[l].b2, x, 256, 128, L, 256, 0, 3);
        }
    }

    // 3) similarity + bias, prob softmax, attn_dir/var
    sim_wmma_k<<<BSZ * 4 * 2, 32, 0, stream>>>(d0p, d1p, lostb, simb);
    softvar_k<<<BSZ * NN, 128, 0, stream>>>(simb, camdirB, probOut, varb);

    // 4) Sinkhorn (LDS-resident, 100 iters) + matching fused
    sinkhorn_k<<<BSZ, 256, 0, stream>>>(simb, bin_score, scoresOut, indOut, msW, idx0W);

    // 5) MLP heads
    heads_k<<<(BSZ * NN + 255) / 256, 256, 0, stream>>>(
        priorPos, odb, camdirB, idx0W, msW, varb,
        p0w, p0b, p1w, p1b, p2w, p2b, p3w, p3b,
        c0w, c0b, c1w, c1b, c2w, c2b,
        posOut, covOut);
}